// HyperIM_55456617726357
// MI455X (gfx1250) — compile-verified
//
#include <hip/hip_runtime.h>
#include <hip/hip_bf16.h>
#include <math.h>

typedef __attribute__((ext_vector_type(2))) float v2f;
typedef __attribute__((ext_vector_type(8))) float v8f;

#define NB   32            // batch
#define SQ   256           // sequence length
#define NL   1024          // labels
#define NH   4             // hyperbolic balls (heads)
#define ND   64            // ball dimension
#define HD   (NH*ND)       // 256
#define F1   128           // hidden of MLP

// ---------------- block-wide sum over 64 threads (2 waves, wave32) ----------
static __device__ __forceinline__ float blk_sum64(float v, float* red) {
  #pragma unroll
  for (int off = 16; off > 0; off >>= 1) v += __shfl_xor(v, off, 32);
  int wid = threadIdx.x >> 5;
  __syncthreads();
  if ((threadIdx.x & 31) == 0) red[wid] = v;
  __syncthreads();
  return red[0] + red[1];
}

// ---------------- Kernel 1: hyperbolic RNN (one block per (b,h) chain) -----
// Each of 64 threads owns one component d of the 64-dim state vector.
// W[h], U[h] are staged transposed in LDS for conflict-free matvec reads.
__global__ void __launch_bounds__(64) hyper_rnn_kernel(
    const int*   __restrict__ xidx,   // [B,S]
    const float* __restrict__ wemb,   // [V,H,D]
    const float* __restrict__ Wm,     // [H,D,D]
    const float* __restrict__ Um,     // [H,D,D]
    const float* __restrict__ bvec,   // [H,D]
    float*       __restrict__ enc) {  // [B,S,H,D]
  __shared__ float WT[ND*ND];
  __shared__ float UT[ND*ND];
  __shared__ float h_lds[ND];
  __shared__ float x_lds[ND];
  __shared__ float red[2];
  const int t = threadIdx.x;
  const int b = blockIdx.x / NH;
  const int h = blockIdx.x % NH;

  const float* Wh = Wm + (size_t)h * ND * ND;
  const float* Uh = Um + (size_t)h * ND * ND;
  #pragma unroll 8
  for (int i = 0; i < ND; ++i) {       // WT[j][i] = W[h][i][j], j = t
    WT[t*ND + i] = Wh[i*ND + t];
    UT[t*ND + i] = Uh[i*ND + t];
  }
  float bb = bvec[h*ND + t];
  float hc = 0.f;
  h_lds[t] = 0.f;
  float bn2 = blk_sum64(bb*bb, red);   // also barriers WT/UT/h_lds into view

  for (int s = 0; s < SQ; ++s) {
    int word = xidx[b*SQ + s];
    float xc = wemb[(size_t)word * HD + h*ND + t];
    x_lds[t] = xc;
    __syncthreads();

    float hn2 = blk_sum64(hc*hc, red);
    float xn2 = blk_sum64(xc*xc, red);

    float whm = 0.f, uxm = 0.f;        // mx = M @ v
    #pragma unroll 8
    for (int j = 0; j < ND; ++j) {
      whm += WT[j*ND + t] * h_lds[j];
      uxm += UT[j*ND + t] * x_lds[j];
    }
    float whn2 = blk_sum64(whm*whm, red);
    float uxn2 = blk_sum64(uxm*uxm, red);

    // mobius_matvec tail: res = tanh(|mx|/|x| * atanh(min(|x|,1-eps))) * mx/|mx|
    float sw, su, wh, ux;
    {
      float mxn = fmaxf(sqrtf(whn2), 1e-15f);
      float xn  = fmaxf(sqrtf(hn2),  1e-15f);
      sw = tanhf(mxn / xn * atanhf(fminf(xn, 1.f - 1e-7f)));
      wh = (whn2 == 0.f) ? 0.f : sw * whm / mxn;
      if (whn2 == 0.f) sw = 0.f;       // ||wh|| = |sw|
    }
    {
      float mxn = fmaxf(sqrtf(uxn2), 1e-15f);
      float xn  = fmaxf(sqrtf(xn2),  1e-15f);
      su = tanhf(mxn / xn * atanhf(fminf(xn, 1.f - 1e-7f)));
      ux = (uxn2 == 0.f) ? 0.f : su * uxm / mxn;
      if (uxn2 == 0.f) su = 0.f;
    }

    // mobius_add(wh, ux): x2/y2 are wave-uniform (unit-scaled vectors)
    float x2 = sw*sw, y2 = su*su;
    float xy = blk_sum64(wh*ux, red);
    float num = (1.f + 2.f*xy + y2) * wh + (1.f - x2) * ux;
    float den = 1.f + 2.f*xy + x2*y2;
    float t1  = num / fmaxf(den, 1e-15f);

    // mobius_add(t1, b)
    float x2b = blk_sum64(t1*t1, red);
    float xyb = blk_sum64(t1*bb, red);
    float num2 = (1.f + 2.f*xyb + bn2) * t1 + (1.f - x2b) * bb;
    float den2 = 1.f + 2.f*xyb + x2b*bn2;
    hc = num2 / fmaxf(den2, 1e-15f);

    h_lds[t] = hc;                     // reads of old h done before blk_sum barriers
    enc[((size_t)(b*SQ + s)) * HD + h*ND + t] = hc;
  }
}

// ---------------- Kernel 2: Poincare distance to all labels + head-sum ------
// One wave per 16x16 (bs, l) tile. f32 WMMA 16x16x4, K=64 per head, 4 heads.
// A 16x4 f32 fragment: lanes 0-15 hold {K=k0,k0+1}, lanes 16-31 hold {K=k0+2,
// k0+3} -> a constant +2-element pointer offset for the high half: load the
// 8-byte fragment directly (no cndmask selects). Row norms cover half the
// columns per lane; one shfl_xor(16) combines the two halves.
__global__ void __launch_bounds__(32) poinc_dist_kernel(
    const float* __restrict__ enc,    // [B*S,H,D]
    const float* __restrict__ lab,    // [L,H,D]
    float*       __restrict__ inter){ // [B,L,S]
  const int bs0  = blockIdx.x * 16;
  const int l0   = blockIdx.y * 16;
  const int lane = threadIdx.x;
  const int m    = lane & 15;
  const bool hi  = lane >= 16;
  const int koff = hi ? 2 : 0;

  float dsum[8];
  #pragma unroll
  for (int r = 0; r < 8; ++r) dsum[r] = 0.f;

  for (int h = 0; h < NH; ++h) {
    const float* Afrag = enc + ((size_t)(bs0 + m) * NH + h) * ND + koff;
    const float* Bfrag = lab + ((size_t)(l0  + m) * NH + h) * ND + koff;
    v8f c = {};
    float a2 = 0.f, b2 = 0.f;          // half-row squared norms
    #pragma unroll
    for (int k0 = 0; k0 < ND; k0 += 4) {
      v2f a  = *(const v2f*)(Afrag + k0);
      v2f bf = *(const v2f*)(Bfrag + k0);
      a2 += a.x*a.x + a.y*a.y;
      b2 += bf.x*bf.x + bf.y*bf.y;
      c = __builtin_amdgcn_wmma_f32_16x16x4_f32(
              false, a, false, bf, (short)0, c, false, false);
    }
    a2 += __shfl_xor(a2, 16, 32);      // combine column halves -> full ||row||^2
    b2 += __shfl_xor(b2, 16, 32);
    #pragma unroll
    for (int r = 0; r < 8; ++r) {      // C layout: VGPR r, M = r + 8*hi, N = m
      int   M    = r + (hi ? 8 : 0);
      float a2m  = __shfl(a2, M, 32);  // lanes M and M+16 hold row-M norm
      float dotv = c[r];
      float c2    = fmaxf(a2m + b2 - 2.f*dotv, 0.f);
      float denom = (1.f - a2m) * (1.f - b2);
      float arg   = 1.f + 2.f * c2 / fmaxf(denom, 1e-15f);
      dsum[r] += acoshf(fmaxf(arg, 1.f + 1e-7f));
    }
  }
  const int bb = bs0 >> 8;             // tile never crosses batch boundary
  const int sb = bs0 & 255;
  #pragma unroll
  for (int r = 0; r < 8; ++r) {
    int M = r + (hi ? 8 : 0);
    inter[(size_t)bb * NL * SQ + (size_t)(l0 + m) * SQ + sb + M] = dsum[r];
  }
}

// ---------------- Kernel 3: MLP  relu(inter @ w1^T + b1) @ w2^T + b2 --------
// One wave per 16 rows of [B*L, S]. A-tile staged in padded LDS, 8 N-tiles.
__global__ void __launch_bounds__(32) mlp_kernel(
    const float* __restrict__ inter,  // [B*L, S]
    const float* __restrict__ w1,     // [F1, S]
    const float* __restrict__ b1,     // [F1]
    const float* __restrict__ w2,     // [1, F1]
    const float* __restrict__ b2,     // [1]
    float*       __restrict__ out) {  // [B*L]
  __shared__ float alds[16 * 260];    // 260-stride pad: conflict-free reads
  const int row0 = blockIdx.x * 16;
  const int lane = threadIdx.x;
  const int m    = lane & 15;
  const bool hi  = lane >= 16;
  const int koff = hi ? 2 : 0;

  for (int idx = lane; idx < 16 * 64; idx += 32) {
    int row = idx >> 6, c4 = idx & 63;
    float4 v = *(const float4*)(inter + (size_t)(row0 + row) * SQ + c4 * 4);
    *(float4*)(&alds[row * 260 + c4 * 4]) = v;
  }
  __syncthreads();

  float rs[8];
  #pragma unroll
  for (int r = 0; r < 8; ++r) rs[r] = 0.f;

  const float* aptr = &alds[m * 260 + koff];
  for (int n0 = 0; n0 < F1 / 16; ++n0) {
    const float* Bfrag = w1 + (size_t)(n0 * 16 + m) * SQ + koff;
    v8f c = {};
    #pragma unroll 16
    for (int k0 = 0; k0 < SQ; k0 += 4) {
      v2f a  = *(const v2f*)(aptr + k0);
      v2f bf = *(const v2f*)(Bfrag + k0);
      c = __builtin_amdgcn_wmma_f32_16x16x4_f32(
              false, a, false, bf, (short)0, c, false, false);
    }
    float b1v = b1[n0 * 16 + m];
    float w2v = w2[n0 * 16 + m];
    #pragma unroll
    for (int r = 0; r < 8; ++r) {
      float hv = fmaxf(c[r] + b1v, 0.f) * w2v;
      hv += __shfl_xor(hv, 1, 32);     // sum over N within each 16-lane half
      hv += __shfl_xor(hv, 2, 32);
      hv += __shfl_xor(hv, 4, 32);
      hv += __shfl_xor(hv, 8, 32);
      rs[r] += hv;
    }
  }
  float b2v = b2[0];
  if (lane == 0) {
    #pragma unroll
    for (int r = 0; r < 8; ++r) out[row0 + r] = rs[r] + b2v;       // M = 0..7
  }
  if (lane == 16) {
    #pragma unroll
    for (int r = 0; r < 8; ++r) out[row0 + 8 + r] = rs[r] + b2v;   // M = 8..15
  }
}

// ---------------- launcher ---------------------------------------------------
extern "C" void kernel_launch(void* const* d_in, const int* in_sizes, int n_in,
                              void* d_out, int out_size, void* d_ws, size_t ws_size,
                              hipStream_t stream) {
  (void)in_sizes; (void)n_in; (void)out_size; (void)ws_size;
  const int*   x    = (const int*)  d_in[0];
  const float* wemb = (const float*)d_in[1];
  const float* lab  = (const float*)d_in[2];
  const float* Wm   = (const float*)d_in[3];
  const float* Um   = (const float*)d_in[4];
  const float* bvec = (const float*)d_in[5];
  const float* w1   = (const float*)d_in[6];
  const float* b1   = (const float*)d_in[7];
  const float* w2   = (const float*)d_in[8];
  const float* b2   = (const float*)d_in[9];
  float* out = (float*)d_out;

  float* enc   = (float*)d_ws;                       // [B*S,H,D]  8 MB
  float* inter = enc + (size_t)NB * SQ * HD;         // [B,L,S]   32 MB

  hyper_rnn_kernel<<<dim3(NB * NH), dim3(64), 0, stream>>>(
      x, wemb, Wm, Um, bvec, enc);
  poinc_dist_kernel<<<dim3(NB * SQ / 16, NL / 16), dim3(32), 0, stream>>>(
      enc, lab, inter);
  mlp_kernel<<<dim3(NB * NL / 16), dim3(32), 0, stream>>>(
      inter, w1, b1, w2, b2, out);
}